// OMP_layer_23270132810495
// MI455X (gfx1250) — compile-verified
//
#include <hip/hip_runtime.h>
#include <math.h>

#define M_DIM 128
#define N_DIM 512
#define BATCH 512
#define KMAX  64

typedef __attribute__((ext_vector_type(2))) float v2f;
typedef __attribute__((ext_vector_type(8))) float v8f;

// ---------------------------------------------------------------------------
// One wave computes a 16x64 strip of C = op(A)*op(B) with V_WMMA_F32_16X16X4_F32
// (4 accumulators; the A fragment is loaded once per K-step and reused by 4
// WMMAs -> 2.5 loads per WMMA instead of 4).
// f32 A frag layout (ISA 7.12.2): lane<16: v0=K0,v1=K1; lane>=16: v0=K2,v1=K3
// (row = lane&15). B frag mirrors. C/D: vgpr p -> (row = p+8*(lane>>4),
// col = lane&15).
// ---------------------------------------------------------------------------
template <typename FA, typename FB>
__device__ __forceinline__ void gemm_strip4(int K, FA loadA, FB loadB, v8f acc[4]) {
  const int lane = threadIdx.x & 31;
  const int hi = lane >> 4;  // 0 or 1
  const int lo = lane & 15;
  for (int k0 = 0; k0 < K; k0 += 4) {
    v2f a;
    a[0] = loadA(lo, k0 + 2 * hi);
    a[1] = loadA(lo, k0 + 2 * hi + 1);
#pragma unroll
    for (int t = 0; t < 4; ++t) {
      v2f b;
      b[0] = loadB(k0 + 2 * hi, 16 * t + lo);
      b[1] = loadB(k0 + 2 * hi + 1, 16 * t + lo);
      acc[t] = __builtin_amdgcn_wmma_f32_16x16x4_f32(
          /*neg_a=*/false, a, /*neg_b=*/false, b,
          /*c_mod=*/(short)0, acc[t], /*reuse_a=*/false, /*reuse_b=*/false);
    }
  }
}

template <typename FS>
__device__ __forceinline__ void store_strip4(const v8f acc[4], FS store) {
  const int lane = threadIdx.x & 31;
  const int hi = lane >> 4;
  const int lo = lane & 15;
#pragma unroll
  for (int t = 0; t < 4; ++t)
#pragma unroll
    for (int p = 0; p < 8; ++p) store(p + 8 * hi, 16 * t + lo, acc[t][p]);
}

// R[m,b] = Y[b,m] - sum_n A[m,n] * X[n,b]   (M=128 x B=512, K=512)
__global__ __launch_bounds__(32) void k_residual(const float* __restrict__ A,
                                                 const float* __restrict__ X,
                                                 const float* __restrict__ Y,
                                                 float* __restrict__ R) {
  const int b0 = blockIdx.x * 64;
  const int m0 = blockIdx.y * 16;
  __builtin_prefetch(&A[m0 * N_DIM], 0, 0);  // global_prefetch_b8
  __builtin_prefetch(&X[b0], 0, 0);
  v8f acc[4] = {};
  gemm_strip4(N_DIM,
      [&](int r, int k) { return A[(m0 + r) * N_DIM + k]; },
      [&](int k, int c) { return X[k * BATCH + b0 + c]; }, acc);
  store_strip4(acc, [&](int r, int c, float v) {
    R[(m0 + r) * BATCH + b0 + c] = Y[(b0 + c) * M_DIM + (m0 + r)] - v;
  });
}

// corr[n,b] = | sum_m A[m,n] * R[m,b] |     (N=512 x B=512, K=128)
__global__ __launch_bounds__(32) void k_corr(const float* __restrict__ A,
                                             const float* __restrict__ R,
                                             float* __restrict__ corr) {
  const int b0 = blockIdx.x * 64;
  const int n0 = blockIdx.y * 16;
  __builtin_prefetch(&A[n0], 0, 0);
  __builtin_prefetch(&R[b0], 0, 0);
  v8f acc[4] = {};
  gemm_strip4(M_DIM,
      [&](int r, int k) { return A[k * N_DIM + n0 + r]; },
      [&](int k, int c) { return R[k * BATCH + b0 + c]; }, acc);
  store_strip4(acc, [&](int r, int c, float v) {
    corr[(n0 + r) * BATCH + b0 + c] = fabsf(v);
  });
}

// Gram[n1,n2] = sum_m A[m,n1] * A[m,n2]     (N=512 x N=512, K=128)
__global__ __launch_bounds__(32) void k_gram(const float* __restrict__ A,
                                             float* __restrict__ Gram) {
  const int n20 = blockIdx.x * 64;
  const int n10 = blockIdx.y * 16;
  v8f acc[4] = {};
  gemm_strip4(M_DIM,
      [&](int r, int k) { return A[k * N_DIM + n10 + r]; },
      [&](int k, int c) { return A[k * N_DIM + n20 + c]; }, acc);
  store_strip4(acc, [&](int r, int c, float v) {
    Gram[(n10 + r) * N_DIM + n20 + c] = v;
  });
}

// rhs[b,n] = sum_m Y[b,m] * A[m,n]          (B=512 x N=512, K=128)
__global__ __launch_bounds__(32) void k_rhs(const float* __restrict__ Y,
                                            const float* __restrict__ A,
                                            float* __restrict__ rhs) {
  const int n0 = blockIdx.x * 64;
  const int b0 = blockIdx.y * 16;
  v8f acc[4] = {};
  gemm_strip4(M_DIM,
      [&](int r, int k) { return Y[(b0 + r) * M_DIM + k]; },
      [&](int k, int c) { return A[k * N_DIM + n0 + c]; }, acc);
  store_strip4(acc, [&](int r, int c, float v) {
    rhs[(b0 + r) * N_DIM + n0 + c] = v;
  });
}

// per-column argmax over n (first-max tie-break, matching jnp.argmax)
__global__ __launch_bounds__(256) void k_argmax(const float* __restrict__ corr,
                                                int* __restrict__ jout) {
  const int b = blockIdx.x;
  __shared__ float sv[256];
  __shared__ int si[256];
  float best = -1.0f;
  int bi = 0;
  for (int n = threadIdx.x; n < N_DIM; n += 256) {
    const float v = corr[n * BATCH + b];
    if (v > best) { best = v; bi = n; }
  }
  sv[threadIdx.x] = best;
  si[threadIdx.x] = bi;
  __syncthreads();
  for (int s = 128; s > 0; s >>= 1) {
    if ((int)threadIdx.x < s) {
      const float v2 = sv[threadIdx.x + s];
      const int i2 = si[threadIdx.x + s];
      if (v2 > sv[threadIdx.x] ||
          (v2 == sv[threadIdx.x] && i2 < si[threadIdx.x])) {
        sv[threadIdx.x] = v2;
        si[threadIdx.x] = i2;
      }
    }
    __syncthreads();
  }
  if (threadIdx.x == 0) jout[b] = si[0];
}

// Snew = S | one-hot(j); also write S output (as float 0/1)
__global__ void k_supdate(const int* __restrict__ Sin,
                          const int* __restrict__ jsel,
                          int* __restrict__ Snew,
                          float* __restrict__ outS) {
  const int i = blockIdx.x * blockDim.x + threadIdx.x;
  if (i >= N_DIM * BATCH) return;
  const int n = i / BATCH;
  const int b = i - n * BATCH;
  const int v = (Sin[i] != 0) || (jsel[b] == n);
  Snew[i] = v;
  outS[i] = v ? 1.0f : 0.0f;
}

// Support-restricted solve per batch: exact equivalent of the reference's
// masked-identity 512x512 solve, on the <=KMAX support submatrix (SPD Gram).
__global__ __launch_bounds__(64) void k_solve(const int* __restrict__ Snew,
                                              const float* __restrict__ Gram,
                                              const float* __restrict__ rhs,
                                              float* __restrict__ outH) {
  const int b = blockIdx.x;
  const int tid = threadIdx.x;
  __shared__ int idx[KMAX];
  __shared__ int kc;
  __shared__ float G[KMAX][KMAX + 1];
  __shared__ float xv[KMAX];

  if (tid == 0) {
    int k = 0;
    for (int n = 0; n < N_DIM && k < KMAX; ++n)
      if (Snew[n * BATCH + b]) idx[k++] = n;
    kc = k;
  }
  __syncthreads();
  const int k = kc;

  for (int t = tid; t < k * k; t += 64) {
    const int i = t / k;
    const int j = t - i * k;
    G[i][j] = Gram[idx[i] * N_DIM + idx[j]];
  }
  for (int i = tid; i < k; i += 64) xv[i] = rhs[b * N_DIM + idx[i]];

  // Gaussian elimination (no pivoting needed: SPD support Gram)
  for (int p = 0; p < k; ++p) {
    __syncthreads();
    const float piv = G[p][p];
    for (int i = p + 1 + tid; i < k; i += 64) {
      const float f = G[i][p] / piv;
      for (int j = p; j < k; ++j) G[i][j] -= f * G[p][j];
      xv[i] -= f * xv[p];
    }
  }
  __syncthreads();
  if (tid == 0) {  // back substitution, k is tiny (~11)
    for (int i = k - 1; i >= 0; --i) {
      float s = xv[i];
      for (int j = i + 1; j < k; ++j) s -= G[i][j] * xv[j];
      xv[i] = s / G[i][i];
    }
  }
  __syncthreads();
  for (int n = tid; n < N_DIM; n += 64) outH[n * BATCH + b] = 0.0f;
  __syncthreads();
  if (tid < k) outH[idx[tid] * BATCH + b] = xv[tid];
}

extern "C" void kernel_launch(void* const* d_in, const int* in_sizes, int n_in,
                              void* d_out, int out_size, void* d_ws, size_t ws_size,
                              hipStream_t stream) {
  const float* X = (const float*)d_in[0];  // [N, B]
  const float* Y = (const float*)d_in[1];  // [B, M]
  const int*   S = (const int*)d_in[2];    // [N, B] bool-as-int
  const float* A = (const float*)d_in[3];  // [M, N]

  float* outH = (float*)d_out;                  // [N, B]
  float* outS = (float*)d_out + N_DIM * BATCH;  // [N, B] as 0/1 floats

  char* ws = (char*)d_ws;
  float* R    = (float*)(ws);                                   // 256 KB
  float* corr = (float*)(ws + 262144);                          // 1 MB
  float* Gram = (float*)(ws + 262144 + 1048576);                // 1 MB
  float* rhsb = (float*)(ws + 262144 + 2 * 1048576);            // 1 MB
  int*   jsel = (int*)  (ws + 262144 + 3 * 1048576);            // 2 KB
  int*   Snew = (int*)  (ws + 262144 + 3 * 1048576 + 4096);     // 1 MB
  (void)in_sizes; (void)n_in; (void)out_size; (void)ws_size;

  const dim3 wave(32);
  k_residual<<<dim3(BATCH / 64, M_DIM / 16), wave, 0, stream>>>(A, X, Y, R);
  k_gram    <<<dim3(N_DIM / 64, N_DIM / 16), wave, 0, stream>>>(A, Gram);
  k_rhs     <<<dim3(N_DIM / 64, BATCH / 16), wave, 0, stream>>>(Y, A, rhsb);
  k_corr    <<<dim3(BATCH / 64, N_DIM / 16), wave, 0, stream>>>(A, R, corr);
  k_argmax  <<<dim3(BATCH), dim3(256), 0, stream>>>(corr, jsel);
  k_supdate <<<dim3((N_DIM * BATCH + 255) / 256), dim3(256), 0, stream>>>(S, jsel, Snew, outS);
  k_solve   <<<dim3(BATCH), dim3(64), 0, stream>>>(Snew, Gram, rhsb, outH);
}